// MillionBucketPyramid_87016037416974
// MI455X (gfx1250) — compile-verified
//
#include <hip/hip_runtime.h>
#include <stdint.h>
#include <stddef.h>

typedef float v2f __attribute__((ext_vector_type(2)));
typedef float v4f __attribute__((ext_vector_type(4)));
typedef float v8f __attribute__((ext_vector_type(8)));

#define BUCKETS 4000000u
#define TLEN 2048u
#define NPOS (32u * 2048u)

__device__ __forceinline__ v2f pick2(v4f v, uint32_t hi) {
    v2f lo; lo.x = v.x; lo.y = v.y;
    v2f hv; hv.x = v.z; hv.y = v.w;
    return hi ? hv : lo;
}

__global__ __launch_bounds__(256) void MillionBucketPyramid_kernel(
    const int*   __restrict__ tokens,   // (32, 2048) values < 1024
    const float* __restrict__ table0,   // (BUCKETS, 4)
    const float* __restrict__ table1,
    const float* __restrict__ table2,
    const float* __restrict__ table3,
    const float* __restrict__ cond_w,   // (8, 8) row-major
    float*       __restrict__ out)      // (32, 2048, 16)
{
    const uint32_t tid  = blockIdx.x * blockDim.x + threadIdx.x;
    const uint32_t wave = tid >> 5;           // global wave id; 16 positions per wave
    const uint32_t lane = threadIdx.x & 31u;
    const uint32_t m    = lane & 15u;         // position within the wave's 16-chunk
    const uint32_t hi   = lane >> 4;          // lane half: selects K {0,1} vs {2,3}
    const uint32_t p    = wave * 16u + m;     // flat (b,t); chunks never cross a row (2048 % 16 == 0)
    const uint32_t b    = p >> 11;
    const uint32_t t    = p & 2047u;

    // ---- rolling hashes (64-bit products, then % 4e6) --------------------
    const int* trow = tokens + (size_t)b * TLEN;
    uint64_t x1 = (t >= 1u) ? (uint64_t)(uint32_t)trow[t - 1u] : 0ull;
    uint64_t x2 = (t >= 2u) ? (uint64_t)(uint32_t)trow[t - 2u] : 0ull;
    uint64_t x3 = (t >= 3u) ? (uint64_t)(uint32_t)trow[t - 3u] : 0ull;
    uint64_t x4 = (t >= 4u) ? (uint64_t)(uint32_t)trow[t - 4u] : 0ull;
    uint64_t x5 = (t >= 5u) ? (uint64_t)(uint32_t)trow[t - 5u] : 0ull;
    uint64_t x6 = (t >= 6u) ? (uint64_t)(uint32_t)trow[t - 6u] : 0ull;
    uint64_t x7 = (t >= 7u) ? (uint64_t)(uint32_t)trow[t - 7u] : 0ull;
    uint64_t x8 = (t >= 8u) ? (uint64_t)(uint32_t)trow[t - 8u] : 0ull;

    const uint64_t P0 = 2654435761ull, P1 = 2246822519ull, P2 = 3266489917ull,
                   P3 = 2028178513ull, P4 = 1220703125ull, P5 = 1610612741ull,
                   P6 = 805306457ull,  P7 = 402653189ull;

    uint64_t h = x1 * P0;
    const uint32_t key0 = (uint32_t)(h % (uint64_t)BUCKETS);
    h ^= x2 * P1;
    const uint32_t key1 = (uint32_t)(h % (uint64_t)BUCKETS);
    h ^= (x3 * P2) ^ (x4 * P3);
    const uint32_t key2 = (uint32_t)(h % (uint64_t)BUCKETS);
    h ^= (x5 * P4) ^ (x6 * P5) ^ (x7 * P6) ^ (x8 * P7);
    const uint32_t key3 = (uint32_t)(h % (uint64_t)BUCKETS);

    // ---- short-embed gathers (16B rows, 16B-aligned) ---------------------
    const v4f e0 = *(const v4f*)(table0 + (size_t)key0 * 4u);
    const v4f e1 = *(const v4f*)(table1 + (size_t)key1 * 4u);

    // ---- conditioning GEMM via V_WMMA_F32_16X16X4_F32 --------------------
    // A(16x4): lane m holds row m; half picks K={0,1} or {2,3}.
    // B(4x16): B[k][n] = cond_w[n][k(+4)] for n<8, else 0 (columns 8..15 dead).
    v2f a1 = pick2(e0, hi);
    v2f a2 = pick2(e1, hi);
    v2f b1; b1.x = 0.0f; b1.y = 0.0f;
    v2f b2 = b1;
    if (m < 8u) {
        const float* wr = cond_w + m * 8u;
        if (hi) { b1.x = wr[2]; b1.y = wr[3]; b2.x = wr[6]; b2.y = wr[7]; }
        else    { b1.x = wr[0]; b1.y = wr[1]; b2.x = wr[4]; b2.y = wr[5]; }
    }
    v8f acc = {};
    acc = __builtin_amdgcn_wmma_f32_16x16x4_f32(false, a1, false, b1, (short)0, acc, false, false);
    acc = __builtin_amdgcn_wmma_f32_16x16x4_f32(false, a2, false, b2, (short)0, acc, false, false);

    // ---- sign bits -> cond_key ------------------------------------------
    // Accumulator layout: VGPR j, lanes 0-15 => (M=j, N=lane); lanes 16-31 => (M=j+8, N=lane-16).
    // Row m sign bits n=0..7 live in mask[m&7], bits [7:0] (m<8) or [23:16] (m>=8).
    const uint32_t mk0 = __builtin_amdgcn_ballot_w32(acc[0] > 0.0f);
    const uint32_t mk1 = __builtin_amdgcn_ballot_w32(acc[1] > 0.0f);
    const uint32_t mk2 = __builtin_amdgcn_ballot_w32(acc[2] > 0.0f);
    const uint32_t mk3 = __builtin_amdgcn_ballot_w32(acc[3] > 0.0f);
    const uint32_t mk4 = __builtin_amdgcn_ballot_w32(acc[4] > 0.0f);
    const uint32_t mk5 = __builtin_amdgcn_ballot_w32(acc[5] > 0.0f);
    const uint32_t mk6 = __builtin_amdgcn_ballot_w32(acc[6] > 0.0f);
    const uint32_t mk7 = __builtin_amdgcn_ballot_w32(acc[7] > 0.0f);

    const uint32_t r = m & 7u;
    uint32_t msel = mk0;
    msel = (r == 1u) ? mk1 : msel;
    msel = (r == 2u) ? mk2 : msel;
    msel = (r == 3u) ? mk3 : msel;
    msel = (r == 4u) ? mk4 : msel;
    msel = (r == 5u) ? mk5 : msel;
    msel = (r == 6u) ? mk6 : msel;
    msel = (r == 7u) ? mk7 : msel;
    const uint32_t sb = (m < 8u) ? (msel & 0xffu) : ((msel >> 16) & 0xffu);

    const uint32_t HP[8] = {2654435761u, 2246822519u, 3266489917u, 2028178513u,
                            1220703125u, 1610612741u, 805306457u,  402653189u};
    uint32_t ck = 0u;
#pragma unroll
    for (int i = 0; i < 8; ++i)
        ck ^= ((sb >> i) & 1u) ? HP[i] : 0u;

    // ---- split the memory work across the lane halves --------------------
    float* orow = out + (size_t)p * 16u;
    if (hi == 0u) {
        *(v4f*)(orow + 0)  = e0;
        *(v4f*)(orow + 4)  = e1;
    } else {
        const uint32_t k2 = (key2 ^ ck) % BUCKETS;
        const uint32_t k3 = (key3 ^ ck) % BUCKETS;
        const v4f e2 = *(const v4f*)(table2 + (size_t)k2 * 4u);
        const v4f e3 = *(const v4f*)(table3 + (size_t)k3 * 4u);
        *(v4f*)(orow + 8)  = e2;
        *(v4f*)(orow + 12) = e3;
    }
}

extern "C" void kernel_launch(void* const* d_in, const int* in_sizes, int n_in,
                              void* d_out, int out_size, void* d_ws, size_t ws_size,
                              hipStream_t stream) {
    (void)in_sizes; (void)n_in; (void)out_size; (void)d_ws; (void)ws_size;
    const int*   tokens = (const int*)  d_in[0];
    const float* table0 = (const float*)d_in[1];
    const float* table1 = (const float*)d_in[2];
    const float* table2 = (const float*)d_in[3];
    const float* table3 = (const float*)d_in[4];
    const float* cond_w = (const float*)d_in[5];
    float*       out    = (float*)d_out;

    // 65536 positions, 16 per wave, 8 waves (256 threads) per block -> 512 blocks.
    const uint32_t nwaves = NPOS / 16u;
    dim3 block(256);
    dim3 grid(nwaves / 8u);
    MillionBucketPyramid_kernel<<<grid, block, 0, stream>>>(
        tokens, table0, table1, table2, table3, cond_w, out);
}